// SourceGCNConvEncoder_26577257628046
// MI455X (gfx1250) — compile-verified
//
#include <hip/hip_runtime.h>
#include <hip/hip_bf16.h>

typedef __attribute__((ext_vector_type(2))) float v2f;
typedef __attribute__((ext_vector_type(8))) float v8f;

static __host__ int cdiv(int a, int b) { return (a + b - 1) / b; }

// ---------------- degree kernels ----------------
__global__ void k_init_deg(int* __restrict__ deg1, int* __restrict__ deg2, int n) {
    int i = blockIdx.x * blockDim.x + threadIdx.x;
    if (i < n) { deg1[i] = 1; deg2[i] = 1; }   // self loops contribute 1
}

__global__ void k_count_deg(const int* __restrict__ rows, const int* __restrict__ cols,
                            int* __restrict__ deg1, int* __restrict__ deg2, int e) {
    int i = blockIdx.x * blockDim.x + threadIdx.x;
    if (i < e) {
        atomicAdd(&deg1[cols[i]], 1);   // layer-1 in-degree (dst = col)
        atomicAdd(&deg2[rows[i]], 1);   // layer-2 (flipped) in-degree (dst = row)
    }
}

// ---------------- WMMA f32 GEMM: Y[n_rows, out] = X[n_rows, K] * W[out, K]^T + bias ----------------
// One wave computes a 16x16 tile of Y, chaining K/4 V_WMMA_F32_16X16X4_F32 ops.
template <int KDIM>
__global__ void k_gemm_wmma_f32(const float* __restrict__ X, const float* __restrict__ W,
                                const float* __restrict__ bias, float* __restrict__ Y,
                                int n_rows, int out_dim) {
    int wid  = (blockIdx.x * blockDim.x + threadIdx.x) >> 5;
    int lane = threadIdx.x & 31;
    int tiles_n = out_dim >> 4;
    int tiles_m = (n_rows + 15) >> 4;
    int tm = wid / tiles_n;
    int tn = wid - tm * tiles_n;
    if (tm >= tiles_m) return;

    int m_l = lane & 15;       // row within tile for A / col within tile for B,C
    int kh  = lane >> 4;       // which K-half (f32 A: lanes 0-15 -> K 0,1 ; lanes 16-31 -> K 2,3)

    int xrow = tm * 16 + m_l; if (xrow >= n_rows) xrow = n_rows - 1;
    int wrow = tn * 16 + m_l;
    const float* xp = X + (size_t)xrow * KDIM + kh * 2;
    const float* wp = W + (size_t)wrow * KDIM + kh * 2;

    // C/D layout: vgpr j, lane L -> row (L>>4)*8 + j, col L&15.  Bias depends only on col.
    float bv = bias[tn * 16 + m_l];
    v8f c;
#pragma unroll
    for (int j = 0; j < 8; ++j) c[j] = bv;

#pragma unroll
    for (int k0 = 0; k0 < KDIM; k0 += 4) {
        v2f a = *(const v2f*)(xp + k0);   // A[m][k0 + 2*kh .. +1]
        v2f b = *(const v2f*)(wp + k0);   // B[k][n] = W[n][k], same lane pattern
        c = __builtin_amdgcn_wmma_f32_16x16x4_f32(
                /*neg_a=*/false, a, /*neg_b=*/false, b,
                /*c_mod=*/(short)0, c, /*reuse_a=*/false, /*reuse_b=*/false);
    }

#pragma unroll
    for (int j = 0; j < 8; ++j) {
        int mr = tm * 16 + kh * 8 + j;
        if (mr < n_rows) Y[(size_t)mr * out_dim + tn * 16 + m_l] = c[j];
    }
}

// ---------------- elementwise copy (self-loop init) ----------------
__global__ void k_copy4(const float4* __restrict__ src, float4* __restrict__ dst, int n4) {
    int i = blockIdx.x * blockDim.x + threadIdx.x;
    if (i < n4) dst[i] = src[i];
}

// ---------------- edge scatter: acc[sidx[i]][:] += src[gidx[i]][:] ----------------
// One wave per edge; 128 channels -> float4 per lane (coalesced 512B gather + 32x4 f32 atomics, L2-resident)
__global__ void k_scatter128(const float* __restrict__ src, float* __restrict__ acc,
                             const int* __restrict__ gidx, const int* __restrict__ sidx, int e) {
    int lane = threadIdx.x & 31;
    int wid  = (blockIdx.x * blockDim.x + threadIdx.x) >> 5;
    int nw   = (gridDim.x * blockDim.x) >> 5;
    for (int i = wid; i < e; i += nw) {
        int g = gidx[i], s = sidx[i];
        float4 v = *(const float4*)(src + (size_t)g * 128 + lane * 4);
        float* d = acc + (size_t)s * 128 + lane * 4;
        atomicAdd(d + 0, v.x); atomicAdd(d + 1, v.y);
        atomicAdd(d + 2, v.z); atomicAdd(d + 3, v.w);
    }
}

// 64 channels -> float2 per lane
__global__ void k_scatter64(const float* __restrict__ src, float* __restrict__ acc,
                            const int* __restrict__ gidx, const int* __restrict__ sidx, int e) {
    int lane = threadIdx.x & 31;
    int wid  = (blockIdx.x * blockDim.x + threadIdx.x) >> 5;
    int nw   = (gridDim.x * blockDim.x) >> 5;
    for (int i = wid; i < e; i += nw) {
        int g = gidx[i], s = sidx[i];
        float2 v = *(const float2*)(src + (size_t)g * 64 + lane * 2);
        float* d = acc + (size_t)s * 64 + lane * 2;
        atomicAdd(d + 0, v.x); atomicAdd(d + 1, v.y);
    }
}

// ---------------- scale by 1/deg (+ optional ReLU), in place, float4 granularity ----------------
__global__ void k_scale(float4* __restrict__ buf, const int* __restrict__ deg,
                        int total4, int c4 /* channels/4 */, int do_relu) {
    int i = blockIdx.x * blockDim.x + threadIdx.x;
    if (i >= total4) return;
    int row = i / c4;
    float inv = 1.0f / (float)deg[row];
    float4 v = buf[i];
    v.x *= inv; v.y *= inv; v.z *= inv; v.w *= inv;
    if (do_relu) {
        v.x = fmaxf(v.x, 0.0f); v.y = fmaxf(v.y, 0.0f);
        v.z = fmaxf(v.z, 0.0f); v.w = fmaxf(v.w, 0.0f);
    }
    buf[i] = v;
}

extern "C" void kernel_launch(void* const* d_in, const int* in_sizes, int n_in,
                              void* d_out, int out_size, void* d_ws, size_t ws_size,
                              hipStream_t stream) {
    const float* x  = (const float*)d_in[0];
    const int*   ei = (const int*)d_in[1];
    const float* W1 = (const float*)d_in[2];
    const float* b1 = (const float*)d_in[3];
    const float* W2 = (const float*)d_in[4];
    const float* b2 = (const float*)d_in[5];

    const int HID = in_sizes[3];              // 128
    const int OUT = in_sizes[5];              // 64
    const int INC = in_sizes[2] / HID;        // 128
    const int n   = in_sizes[0] / INC;        // 50000
    const int e   = in_sizes[1] / 2;          // 800000
    const int* rows = ei;
    const int* cols = ei + e;

    // workspace: deg1 | deg2 | Y (n*HID) | ACC (n*HID)
    int*   deg1 = (int*)d_ws;
    int*   deg2 = deg1 + n;
    float* Y    = (float*)(deg2 + n);
    float* ACC  = Y + (size_t)n * HID;
    float* out  = (float*)d_out;

    // degrees (self loops = init 1)
    k_init_deg<<<cdiv(n, 256), 256, 0, stream>>>(deg1, deg2, n);
    k_count_deg<<<cdiv(e, 256), 256, 0, stream>>>(rows, cols, deg1, deg2, e);

    // ---- layer 1 ----
    {   // Y = x @ W1^T + b1   (K = INC = 128)
        int waves = cdiv(n, 16) * (HID / 16);
        k_gemm_wmma_f32<128><<<cdiv(waves, 8), 256, 0, stream>>>(x, W1, b1, Y, n, HID);
    }
    {   // ACC = Y (self loop), then ACC[col] += Y[row], then h = relu(ACC/deg1) in place
        int n4 = n * HID / 4;
        k_copy4<<<cdiv(n4, 256), 256, 0, stream>>>((const float4*)Y, (float4*)ACC, n4);
        k_scatter128<<<2048, 256, 0, stream>>>(Y, ACC, rows, cols, e);
        k_scale<<<cdiv(n4, 256), 256, 0, stream>>>((float4*)ACC, deg1, n4, HID / 4, 1);
    }

    // ---- layer 2 (edges flipped: gather from col, scatter to row) ----
    {   // Y2 = h @ W2^T + b2 (reuse Y buffer; h lives in ACC)
        int waves = cdiv(n, 16) * (OUT / 16);
        k_gemm_wmma_f32<128><<<cdiv(waves, 8), 256, 0, stream>>>(ACC, W2, b2, Y, n, OUT);
    }
    {
        int n4 = n * OUT / 4;
        k_copy4<<<cdiv(n4, 256), 256, 0, stream>>>((const float4*)Y, (float4*)out, n4);
        k_scatter64<<<2048, 256, 0, stream>>>(Y, out, cols, rows, e);
        k_scale<<<cdiv(n4, 256), 256, 0, stream>>>((float4*)out, deg2, n4, OUT / 4, 0);
    }
}